// AttentionBlock_5772436046107
// MI455X (gfx1250) — compile-verified
//
#include <hip/hip_runtime.h>
#include <hip/hip_bf16.h>
#include <math.h>

typedef __attribute__((ext_vector_type(2))) float v2f;
typedef __attribute__((ext_vector_type(8))) float v8f;

// ---------------------------------------------------------------------------
// Kernel 1: 31x31/stride-16 box average pool with edge-corrected counts.
// One block per (b,c) plane (1024 planes), 256 threads.
// Separable: horizontal window sums per row -> vertical sums per pooled row.
// ---------------------------------------------------------------------------
__global__ __launch_bounds__(256) void pool_kernel(const float* __restrict__ x,
                                                   float* __restrict__ p) {
    const int plane = blockIdx.x;                   // b*128 + c
    const float* xp = x + (size_t)plane * 256 * 256;

    __shared__ float tile[16 * 256];                // 16 rows staged at a time
    __shared__ float rowsum[256][17];               // padded: conflict-free

    const int t       = threadIdx.x;
    const int r_local = t >> 4;                     // 0..15
    const int pw      = t & 15;                     // 0..15

    // horizontal window for this pw: [cstart, cend] (end never clips: 15*16+15=255)
    const int cstart = (pw == 0) ? 0 : (pw * 16 - 15);
    const int cend   = pw * 16 + 15;

    for (int rb = 0; rb < 16; ++rb) {
        // stage rows rb*16 .. rb*16+15 (4096 floats) coalesced via float4
        const float4* src = (const float4*)(xp + rb * 16 * 256);
        float4* dst = (float4*)tile;
        #pragma unroll
        for (int i = 0; i < 4; ++i) {
            dst[t + i * 256] = src[t + i * 256];
        }
        __syncthreads();

        float hsum = 0.f;
        const float* row = tile + r_local * 256;
        for (int c = cstart; c <= cend; ++c) hsum += row[c];
        rowsum[rb * 16 + r_local][pw] = hsum;
        __syncthreads();
    }

    // vertical reduction: thread t handles (ph = r_local, pw)
    const int ph     = r_local;
    const int rstart = (ph == 0) ? 0 : (ph * 16 - 15);
    const int rend   = ph * 16 + 15;
    float vsum = 0.f;
    for (int r = rstart; r <= rend; ++r) vsum += rowsum[r][pw];

    const float count = (float)((rend - rstart + 1) * (cend - cstart + 1));
    p[(size_t)plane * 256 + ph * 16 + pw] = vsum / count;
}

// ---------------------------------------------------------------------------
// Kernel 2: per-pixel channel MLP  s = sigmoid(W2 @ relu(W1 @ p + b1) + b2)
// 2048 pixels total; one wave (32 threads) per 16-pixel M-tile -> 128 blocks.
// Full fp32 via V_WMMA_F32_16X16X4_F32.
//   A 16x4 layout: lanes 0-15 hold M=lane {K0,K1}; lanes 16-31 hold {K2,K3}.
//   B 4x16 layout: VGPR0 = rows {K0 | K2}, VGPR1 = rows {K1 | K3} per half-wave.
//   C/D: VGPR v -> M = v + (lane>=16 ? 8 : 0), N = lane & 15.
// ---------------------------------------------------------------------------
__global__ __launch_bounds__(32) void mlp_kernel(const float* __restrict__ p,
                                                 const float* __restrict__ w1,
                                                 const float* __restrict__ b1,
                                                 const float* __restrict__ w2,
                                                 const float* __restrict__ b2,
                                                 float* __restrict__ sout) {
    const int tile    = blockIdx.x;                 // 0..127
    const int lane    = threadIdx.x;                // 0..31
    const int b       = tile >> 4;                  // 16 tiles per batch image
    const int pixbase = (tile & 15) * 16;           // pixel offset within batch
    const int mrow    = lane & 15;                  // M (or N) index for this lane
    const int khalf   = (lane >> 4) * 2;            // 0 or 2 : K sub-pair
    const int mbase   = (lane >> 4) * 8;            // M offset in C/D layout

    // p[b][c][ph][pw]: address = (b*128 + c)*256 + pix_in_batch
    const float* pb = p + (size_t)b * 128 * 256;

    // ---------------- GEMM1: H[16][32] = P[16][128] x W1^T, relu(+b1) -------
    v8f acc0 = {};  // N = 0..15
    v8f acc1 = {};  // N = 16..31
    for (int k0 = 0; k0 < 128; k0 += 4) {
        v2f a;
        const float* pa = pb + (size_t)(k0 + khalf) * 256 + pixbase + mrow;
        a.x = pa[0];
        a.y = pa[256];
        v2f bA, bB;                                 // B[k][n] = w1[n][k]
        const float* w1n0 = w1 + (size_t)mrow * 128 + k0 + khalf;
        const float* w1n1 = w1 + (size_t)(mrow + 16) * 128 + k0 + khalf;
        bA.x = w1n0[0]; bA.y = w1n0[1];
        bB.x = w1n1[0]; bB.y = w1n1[1];
        acc0 = __builtin_amdgcn_wmma_f32_16x16x4_f32(false, a, false, bA,
                                                     (short)0, acc0, false, false);
        acc1 = __builtin_amdgcn_wmma_f32_16x16x4_f32(false, a, false, bB,
                                                     (short)0, acc1, false, false);
    }

    // bias + relu, stage H through LDS to re-stripe for GEMM2's A layout
    __shared__ float Hs[16][33];
    {
        const float bias0 = b1[mrow];
        const float bias1 = b1[mrow + 16];
        #pragma unroll
        for (int v = 0; v < 8; ++v) {
            Hs[mbase + v][mrow]      = fmaxf(acc0[v] + bias0, 0.f);
            Hs[mbase + v][mrow + 16] = fmaxf(acc1[v] + bias1, 0.f);
        }
    }
    __syncthreads();

    // ---------------- GEMM2: S[16][128] = H[16][32] x W2^T, sigmoid(+b2) ----
    v8f acc[8];
    #pragma unroll
    for (int nt = 0; nt < 8; ++nt) acc[nt] = (v8f){};

    for (int k0 = 0; k0 < 32; k0 += 4) {
        v2f a;
        a.x = Hs[mrow][k0 + khalf];
        a.y = Hs[mrow][k0 + khalf + 1];
        #pragma unroll
        for (int nt = 0; nt < 8; ++nt) {
            v2f bb;                                 // B[k][n] = w2[n][k]
            const float* w2n = w2 + (size_t)(nt * 16 + mrow) * 32 + k0 + khalf;
            bb.x = w2n[0]; bb.y = w2n[1];
            acc[nt] = __builtin_amdgcn_wmma_f32_16x16x4_f32(false, a, false, bb,
                                                            (short)0, acc[nt],
                                                            false, false);
        }
    }

    // bias + sigmoid, scatter to s[b][c][ph][pw]
    #pragma unroll
    for (int nt = 0; nt < 8; ++nt) {
        const int   c    = nt * 16 + mrow;
        const float bias = b2[c];
        float* sp = sout + ((size_t)b * 128 + c) * 256 + pixbase + mbase;
        #pragma unroll
        for (int v = 0; v < 8; ++v) {
            const float z = acc[nt][v] + bias;
            sp[v] = 1.f / (1.f + __expf(-z));
        }
    }
}

// ---------------------------------------------------------------------------
// Kernel 3: out = x * upsample16(s). Pure streaming float4 pass.
// Each float4 lies inside one 16-column group -> single s scalar per vector.
// ---------------------------------------------------------------------------
__global__ __launch_bounds__(256) void scale_kernel(const float* __restrict__ x,
                                                    const float* __restrict__ s,
                                                    float* __restrict__ out) {
    const size_t i = (size_t)blockIdx.x * blockDim.x + threadIdx.x;  // float4 idx
    const size_t e = i * 4;                      // element index
    const int w  = (int)(e & 255);
    const int h  = (int)((e >> 8) & 255);
    const int cb = (int)(e >> 16);               // b*128 + c

    const float sv = s[(size_t)cb * 256 + (h >> 4) * 16 + (w >> 4)];
    const float4 xv = ((const float4*)x)[i];
    float4 o;
    o.x = xv.x * sv; o.y = xv.y * sv; o.z = xv.z * sv; o.w = xv.w * sv;
    ((float4*)out)[i] = o;
}

// ---------------------------------------------------------------------------
extern "C" void kernel_launch(void* const* d_in, const int* in_sizes, int n_in,
                              void* d_out, int out_size, void* d_ws, size_t ws_size,
                              hipStream_t stream) {
    const float* x  = (const float*)d_in[0];   // (8,128,256,256)
    const float* w1 = (const float*)d_in[1];   // (32,128)
    const float* b1 = (const float*)d_in[2];   // (32)
    const float* w2 = (const float*)d_in[3];   // (128,32)
    const float* b2 = (const float*)d_in[4];   // (128)
    float* out = (float*)d_out;

    float* p = (float*)d_ws;                   // (8,128,16,16) = 1 MB
    float* s = p + (size_t)8 * 128 * 16 * 16;  // (8,128,16,16) = 1 MB

    // Pass 1: pooled averages (reads x once)
    pool_kernel<<<8 * 128, 256, 0, stream>>>(x, p);

    // Pass 2: WMMA fp32 MLP (tiny)
    mlp_kernel<<<128, 32, 0, stream>>>(p, w1, b1, w2, b2, s);

    // Pass 3: elementwise scale (reads x once, writes out once)
    const size_t nvec = (size_t)8 * 128 * 256 * 256 / 4;   // 16,777,216 float4
    scale_kernel<<<(unsigned)(nvec / 256), 256, 0, stream>>>(x, s, out);
}